// DSSnetwork_Atten_627065225444
// MI455X (gfx1250) — compile-verified
//
#include <hip/hip_runtime.h>
#include <hip/hip_bf16.h>

#define G_ 128
#define S_ 32
#define D_ 128
#define L_ 3
#define H_ 4
#define NTASK_ 10
#define N_ 32
#define T_ (G_ * S_ * N_)      // 131072
#define TO_ (G_ * N_)          // 4096
#define E_ 1048576
#define EO_ 16384
#define HD_ 32

typedef float v2f __attribute__((ext_vector_type(2)));
typedef float v8f __attribute__((ext_vector_type(8)));

__device__ __forceinline__ v8f wmma_f32_k4(v2f a, v2f b, v8f c) {
  // V_WMMA_F32_16X16X4_F32: full-precision fp32 matrix op (wave32)
  return __builtin_amdgcn_wmma_f32_16x16x4_f32(false, a, false, b, (short)0, c,
                                               false, false);
}

// out[M x 128] = A @ W1 (+ Agg @ W2 if dual) + bias.  W row-major 128x128.
// One wave computes a full 16x128 output row-block: A tile loaded ONCE per
// K-step (b64), 8 independent accumulators so consecutive v_wmma have no
// D->A/B hazard. A layout per ISA 16x4 f32: lanes 0-15 hold K=k0,k0+1,
// lanes 16-31 hold K=k0+2,k0+3; C/D vgpr r = row (r + 8*half).
__global__ void gemm_wmma_kernel(const float* __restrict__ A,
                                 const float* __restrict__ W1,
                                 const float* __restrict__ Agg,
                                 const float* __restrict__ W2,
                                 const float* __restrict__ bias,
                                 float* __restrict__ out, int M, int dual) {
  int wave = blockIdx.x * (blockDim.x >> 5) + (threadIdx.x >> 5);
  if (wave * 16 >= M) return;
  int lane = threadIdx.x & 31;
  int half = lane >> 4;
  int lr = lane & 15;
  int row = wave * 16 + lr;

  v8f c[8];
#pragma unroll
  for (int t = 0; t < 8; ++t) {
    float bv = bias[t * 16 + lr];
#pragma unroll
    for (int r = 0; r < 8; ++r) c[t][r] = bv;
  }

  const float* Arow = A + (size_t)row * D_;
  for (int k0 = 0; k0 < D_; k0 += 4) {
    int ka = k0 + 2 * half;
    v2f a;
    a.x = Arow[ka];
    a.y = Arow[ka + 1];
    const float* Wk = W1 + (size_t)ka * D_ + lr;
#pragma unroll
    for (int t = 0; t < 8; ++t) {
      v2f b;
      b.x = Wk[t * 16];
      b.y = Wk[t * 16 + D_];
      c[t] = wmma_f32_k4(a, b, c[t]);
    }
  }
  if (dual) {
    const float* Grow = Agg + (size_t)row * D_;
    for (int k0 = 0; k0 < D_; k0 += 4) {
      int ka = k0 + 2 * half;
      v2f a;
      a.x = Grow[ka];
      a.y = Grow[ka + 1];
      const float* Wk = W2 + (size_t)ka * D_ + lr;
#pragma unroll
      for (int t = 0; t < 8; ++t) {
        v2f b;
        b.x = Wk[t * 16];
        b.y = Wk[t * 16 + D_];
        c[t] = wmma_f32_k4(a, b, c[t]);
      }
    }
  }
#pragma unroll
  for (int t = 0; t < 8; ++t) {
#pragma unroll
    for (int r = 0; r < 8; ++r) {
      int ro = wave * 16 + half * 8 + r;
      out[(size_t)ro * D_ + t * 16 + lr] = c[t][r];
    }
  }
}

// agg[dst] += x[src] (per-feature fp32 atomics), one thread = (edge, float4)
__global__ void scatter_add_kernel(const float* __restrict__ x,
                                   const int* __restrict__ src,
                                   const int* __restrict__ dst,
                                   float* __restrict__ agg, int nE) {
  long long tid = (long long)blockIdx.x * blockDim.x + threadIdx.x;
  int e = (int)(tid >> 5);
  if (e >= nE) return;
  int c = (int)(tid & 31);
  int s = src[e], d0 = dst[e];
  float4 v = *(const float4*)(x + (size_t)s * D_ + c * 4);
  float* o = agg + (size_t)d0 * D_ + c * 4;
  atomicAdd(o + 0, v.x);
  atomicAdd(o + 1, v.y);
  atomicAdd(o + 2, v.z);
  atomicAdd(o + 3, v.w);
}

// Flat coalesced per-column sum/sumsq accumulation. Grid-stride is a multiple
// of 128, so every element a thread touches lives in column (tid & 127).
// sums[0:128]=sum, sums[128:256]=sumsq (must be zeroed before launch).
__global__ void colstats_accum_kernel(const float* __restrict__ h, long long total,
                                      float* __restrict__ sums) {
  __shared__ float ss[256], s2[256];
  long long stride = (long long)gridDim.x * blockDim.x;
  float a = 0.0f, b = 0.0f;
  for (long long i = (long long)blockIdx.x * blockDim.x + threadIdx.x; i < total;
       i += stride) {
    float v = h[i];
    a += v;
    b += v * v;
  }
  ss[threadIdx.x] = a;
  s2[threadIdx.x] = b;
  __syncthreads();
  if (threadIdx.x < 128) {
    atomicAdd(&sums[threadIdx.x], ss[threadIdx.x] + ss[threadIdx.x + 128]);
    atomicAdd(&sums[128 + threadIdx.x], s2[threadIdx.x] + s2[threadIdx.x + 128]);
  }
}

__global__ void colstats_final_kernel(float* __restrict__ sums, float invM) {
  int d = threadIdx.x;  // 128 threads
  float mu = sums[d] * invM;
  float va = sums[128 + d] * invM - mu * mu;
  sums[d] = mu;
  sums[128 + d] = va;
}

// sub[g,s,d] = mean over n of x[g,s,n,d]
__global__ void sub_mean_kernel(const float* __restrict__ x, float* __restrict__ sub) {
  int idx = blockIdx.x * blockDim.x + threadIdx.x;  // G*S*D
  if (idx >= G_ * S_ * D_) return;
  int d = idx & 127;
  int gs = idx >> 7;
  const float* p = x + (size_t)gs * N_ * D_ + d;
  float s = 0.0f;
#pragma unroll 8
  for (int n = 0; n < N_; ++n) s += p[(size_t)n * D_];
  sub[idx] = s * (1.0f / N_);
}

// scores[g,h,i,j] = scale * dot(q[g,i,h,:], k[g,j,h,:])
__global__ void scores_kernel(const float* __restrict__ q, const float* __restrict__ k,
                              float* __restrict__ scores) {
  int idx = blockIdx.x * blockDim.x + threadIdx.x;  // G*H*S*S
  if (idx >= G_ * H_ * S_ * S_) return;
  int j = idx & 31, i = (idx >> 5) & 31, h = (idx >> 10) & 3, g = idx >> 12;
  const float* qp = q + (size_t)(g * S_ + i) * D_ + h * HD_;
  const float* kp = k + (size_t)(g * S_ + j) * D_ + h * HD_;
  float s = 0.0f;
#pragma unroll 8
  for (int d = 0; d < HD_; ++d) s += qp[d] * kp[d];
  scores[idx] = s * 0.17677669529663687f;  // 1/sqrt(32)
}

// softmax over rows of 32 — one wave32 per row, shuffle reductions
__global__ void softmax_rows_kernel(float* __restrict__ scores, int nRows) {
  int wave = blockIdx.x * (blockDim.x >> 5) + (threadIdx.x >> 5);
  if (wave >= nRows) return;
  int lane = threadIdx.x & 31;
  float* rowp = scores + (size_t)wave * 32;
  float v = rowp[lane];
  float m = v;
  for (int off = 16; off > 0; off >>= 1) m = fmaxf(m, __shfl_xor(m, off, 32));
  float e = __expf(v - m);
  float s = e;
  for (int off = 16; off > 0; off >>= 1) s += __shfl_xor(s, off, 32);
  rowp[lane] = e / s;
}

__global__ void headmean_kernel(const float* __restrict__ probs, float* __restrict__ attn) {
  int idx = blockIdx.x * blockDim.x + threadIdx.x;  // G*S*S
  if (idx >= G_ * S_ * S_) return;
  int ij = idx & 1023;
  int g = idx >> 10;
  const float* p = probs + (size_t)g * (H_ * S_ * S_) + ij;
  attn[idx] = 0.25f * (p[0] + p[1024] + p[2048] + p[3072]);
}

// x_atten[g,n,d] = sum_s attn[g,s,n] * x[g,s,n,d]
__global__ void xatten_kernel(const float* __restrict__ x, const float* __restrict__ attn,
                              float* __restrict__ xatt) {
  int idx = blockIdx.x * blockDim.x + threadIdx.x;  // TO*D
  if (idx >= TO_ * D_) return;
  int d = idx & 127;
  int gn = idx >> 7;
  int n = gn & 31;
  int g = gn >> 5;
  const float* xp = x + ((size_t)g * S_ * N_ + n) * D_ + d;  // stride over s = N*D
  const float* ap = attn + (size_t)g * S_ * S_ + n;          // stride over s = S
  float s = 0.0f;
#pragma unroll 8
  for (int ss = 0; ss < S_; ++ss) s += ap[ss * S_] * xp[(size_t)ss * N_ * D_];
  xatt[idx] = s;
}

// x_next = relu(BN(h1) + BN(h2)[broadcast over s]), in-place on h1
__global__ void combine_kernel(float* __restrict__ h1, const float* __restrict__ h2,
                               const float* __restrict__ stats,
                               const float* __restrict__ g1, const float* __restrict__ be1,
                               const float* __restrict__ g2, const float* __restrict__ be2) {
  int idx = blockIdx.x * blockDim.x + threadIdx.x;  // T*D
  if (idx >= T_ * D_) return;
  int d = idx & 127;
  int t = idx >> 7;
  int n = t & 31;
  int g = t >> 10;
  int go = g * N_ + n;
  float a = g1[d] * (h1[idx] - stats[d]) * rsqrtf(stats[128 + d] + 1e-5f) + be1[d];
  float b = g2[d] * (h2[(size_t)go * D_ + d] - stats[256 + d]) *
                rsqrtf(stats[384 + d] + 1e-5f) + be2[d];
  h1[idx] = fmaxf(a + b, 0.0f);
}

__global__ void pool_kernel(const float* __restrict__ x, float* __restrict__ hg) {
  int idx = blockIdx.x * blockDim.x + threadIdx.x;  // G*D (coalesced across d)
  if (idx >= G_ * D_) return;
  int d = idx & 127;
  int g = idx >> 7;
  const float* p = x + (size_t)g * S_ * N_ * D_ + d;
  float s = 0.0f;
  for (int r = 0; r < S_ * N_; ++r) s += p[(size_t)r * D_];
  hg[idx] = s * (1.0f / (S_ * N_));
}

__global__ void mlp1_kernel(const float* __restrict__ hg, const float* __restrict__ Wf1,
                            const float* __restrict__ bf1, float* __restrict__ f1) {
  int idx = blockIdx.x * blockDim.x + threadIdx.x;  // G * 2D
  if (idx >= G_ * 2 * D_) return;
  int j = idx & 255;
  int g = idx >> 8;
  float s = bf1[j];
  const float* h = hg + (size_t)g * D_;
#pragma unroll 8
  for (int d = 0; d < D_; ++d) s += h[d] * Wf1[(size_t)d * (2 * D_) + j];
  f1[idx] = fmaxf(s, 0.0f);
}

__global__ void mlp2_kernel(const float* __restrict__ f1, const float* __restrict__ Wf2,
                            const float* __restrict__ bf2, float* __restrict__ out) {
  int idx = blockIdx.x * blockDim.x + threadIdx.x;  // G * NTASK
  if (idx >= G_ * NTASK_) return;
  int t = idx % NTASK_;
  int g = idx / NTASK_;
  float s = bf2[t];
  const float* f = f1 + (size_t)g * 2 * D_;
  for (int j = 0; j < 2 * D_; ++j) s += f[j] * Wf2[(size_t)j * NTASK_ + t];
  out[idx] = s;
}

__global__ void heatmap_kernel(const float* __restrict__ attn, float* __restrict__ out) {
  int idx = blockIdx.x * blockDim.x + threadIdx.x;
  if (idx < S_ * S_) out[G_ * NTASK_ + idx] = attn[(size_t)(G_ - 1) * S_ * S_ + idx];
}

extern "C" void kernel_launch(void* const* d_in, const int* in_sizes, int n_in,
                              void* d_out, int out_size, void* d_ws, size_t ws_size,
                              hipStream_t stream) {
  (void)in_sizes; (void)n_in; (void)out_size; (void)ws_size;
  const float* x_in = (const float*)d_in[0];
  const int* ei = (const int*)d_in[1];
  const int* oei = (const int*)d_in[2];
  const float* Wr1 = (const float*)d_in[3];
  const float* Wn1 = (const float*)d_in[4];
  const float* b1 = (const float*)d_in[5];
  const float* g1 = (const float*)d_in[6];
  const float* be1 = (const float*)d_in[7];
  const float* Wr2 = (const float*)d_in[8];
  const float* Wn2 = (const float*)d_in[9];
  const float* b2 = (const float*)d_in[10];
  const float* g2 = (const float*)d_in[11];
  const float* be2 = (const float*)d_in[12];
  const float* Wq = (const float*)d_in[13];
  const float* bq = (const float*)d_in[14];
  const float* Wk = (const float*)d_in[15];
  const float* bk = (const float*)d_in[16];
  const float* Wf1 = (const float*)d_in[17];
  const float* bf1 = (const float*)d_in[18];
  const float* Wf2 = (const float*)d_in[19];
  const float* bf2 = (const float*)d_in[20];
  float* out = (float*)d_out;

  const int* src = ei;
  const int* dstp = ei + E_;
  const int* osrc = oei;
  const int* odst = oei + EO_;

  float* ws = (float*)d_ws;
  const size_t TD = (size_t)T_ * D_;
  float* hA = ws;
  float* hB = ws + TD;
  float* agg = ws + 2 * TD;
  float* sub = ws + 3 * TD;
  float* q = sub + (size_t)G_ * S_ * D_;
  float* kk = q + (size_t)G_ * S_ * D_;
  float* probs = kk + (size_t)G_ * S_ * D_;
  float* attn = probs + (size_t)G_ * H_ * S_ * S_;
  float* xatt = attn + (size_t)G_ * S_ * S_;
  float* h2 = xatt + (size_t)TO_ * D_;
  float* stats = h2 + (size_t)TO_ * D_;  // sum1/mu1, sumsq1/var1, sum2/mu2, sumsq2/var2
  float* hg = stats + 512;
  float* f1 = hg + (size_t)G_ * D_;

  const float* xcur = x_in;
  for (int i = 0; i < L_; ++i) {
    float* hnext = (i & 1) ? hB : hA;
    const float* Wr1i = Wr1 + (size_t)i * D_ * D_;
    const float* Wn1i = Wn1 + (size_t)i * D_ * D_;
    const float* b1i = b1 + (size_t)i * D_;
    const float* Wr2i = Wr2 + (size_t)i * D_ * D_;
    const float* Wn2i = Wn2 + (size_t)i * D_ * D_;
    const float* b2i = b2 + (size_t)i * D_;
    const float* Wqi = Wq + (size_t)i * D_ * D_;
    const float* bqi = bq + (size_t)i * D_;
    const float* Wki = Wk + (size_t)i * D_ * D_;
    const float* bki = bk + (size_t)i * D_;

    // conv1 over subgraph graph
    hipMemsetAsync(agg, 0, TD * sizeof(float), stream);
    hipMemsetAsync(stats, 0, 512 * sizeof(float), stream);
    scatter_add_kernel<<<((size_t)E_ * 32 + 255) / 256, 256, 0, stream>>>(
        xcur, src, dstp, agg, E_);
    gemm_wmma_kernel<<<T_ / 16 / 4, 128, 0, stream>>>(xcur, Wr1i, agg, Wn1i, b1i,
                                                      hnext, T_, 1);
    colstats_accum_kernel<<<1024, 256, 0, stream>>>(hnext, (long long)TD, stats);
    colstats_final_kernel<<<1, 128, 0, stream>>>(stats, 1.0f / (float)T_);

    // attention path
    sub_mean_kernel<<<(G_ * S_ * D_) / 256, 256, 0, stream>>>(xcur, sub);
    gemm_wmma_kernel<<<G_ * S_ / 16 / 4, 128, 0, stream>>>(sub, Wqi, nullptr, nullptr,
                                                           bqi, q, G_ * S_, 0);
    gemm_wmma_kernel<<<G_ * S_ / 16 / 4, 128, 0, stream>>>(sub, Wki, nullptr, nullptr,
                                                           bki, kk, G_ * S_, 0);
    scores_kernel<<<(G_ * H_ * S_ * S_) / 256, 256, 0, stream>>>(q, kk, probs);
    softmax_rows_kernel<<<(G_ * H_ * S_ + 7) / 8, 256, 0, stream>>>(probs, G_ * H_ * S_);
    headmean_kernel<<<(G_ * S_ * S_) / 256, 256, 0, stream>>>(probs, attn);
    xatten_kernel<<<(TO_ * D_) / 256, 256, 0, stream>>>(xcur, attn, xatt);

    // conv2 over original graph
    hipMemsetAsync(agg, 0, (size_t)TO_ * D_ * sizeof(float), stream);
    scatter_add_kernel<<<((size_t)EO_ * 32 + 255) / 256, 256, 0, stream>>>(
        xatt, osrc, odst, agg, EO_);
    gemm_wmma_kernel<<<TO_ / 16 / 4, 128, 0, stream>>>(xatt, Wr2i, agg, Wn2i, b2i,
                                                       h2, TO_, 1);
    colstats_accum_kernel<<<256, 256, 0, stream>>>(h2, (long long)TO_ * D_, stats + 256);
    colstats_final_kernel<<<1, 128, 0, stream>>>(stats + 256, 1.0f / (float)TO_);

    // fused BN + BN + broadcast + relu (in place on hnext)
    combine_kernel<<<(int)(TD / 256), 256, 0, stream>>>(hnext, h2, stats, g1 + i * D_,
                                                        be1 + i * D_, g2 + i * D_,
                                                        be2 + i * D_);
    xcur = hnext;
  }

  pool_kernel<<<(G_ * D_) / 256, 256, 0, stream>>>(xcur, hg);
  mlp1_kernel<<<(G_ * 2 * D_) / 256, 256, 0, stream>>>(hg, Wf1, bf1, f1);
  mlp2_kernel<<<(G_ * NTASK_ + 255) / 256, 256, 0, stream>>>(f1, Wf2, bf2, out);
  heatmap_kernel<<<(S_ * S_ + 255) / 256, 256, 0, stream>>>(attn, out);
}